// EnhancedTrajectoryPredictor_30975304139251
// MI455X (gfx1250) — compile-verified
//
#include <hip/hip_runtime.h>
#include <hip/hip_bf16.h>
#include <stdint.h>

// ---------------------------------------------------------------------------
// GNN forward, algebraically restructured:
//   pairs@W  ->  h@W_top (per-node) + h@W_bot (per-node)
//   sum_j att * (relu(pre)@Wm2)  ->  (sum_j att * relu(pre)) @ Wm2
// leaving per-node GEMMs (WMMA f32 16x16x4) + a flash-style pairwise pass.
// ---------------------------------------------------------------------------

typedef __attribute__((ext_vector_type(2))) float v2f;
typedef __attribute__((ext_vector_type(8))) float v8f;

#define GNN_B  8
#define GNN_N  128
#define GNN_T  4
#define GNN_F  64
#define GNN_H  128
#define GNN_IN 45
#define GNN_BT (GNN_B * GNN_T)
#define GNN_M  (GNN_BT * GNN_N)   // 4096 rows in node-major layout

// ---------------------------------------------------------------------------
// CDNA5 async global->LDS copy (ASYNCcnt-tracked), 16 bytes per lane.
// ---------------------------------------------------------------------------
__device__ __forceinline__ void async_copy_b128(uint32_t lds_off, const void* gptr)
{
    asm volatile("global_load_async_to_lds_b128 %0, %1, off"
                 :: "v"(lds_off), "v"((uint64_t)(uintptr_t)gptr)
                 : "memory");
}
__device__ __forceinline__ void wait_asynccnt0()
{
    asm volatile("s_wait_asynccnt 0x0" ::: "memory");
}

// ---------------------------------------------------------------------------
// f32 WMMA GEMM: D = act( A(MxK) @ B(KxN) + bias + Cadd ), K compile-time.
// Block = 256 thr (8 waves). All waves share one 16-col N strip of B, staged
// in LDS in per-lane fragment order so the inner loop is
//   ds_load_b64 (B frag) + 2x global_load_b64 (A frags) + 2x v_wmma.
// Each wave computes 2 M-tiles (32 rows); block covers 256 rows.
// Lane layout per CDNA5 ISA 7.12.2:
//   A frag (16x4):  M = lane%16, v0/v1 = K (0/1 lanes<16, 2/3 lanes>=16)
//   B frag (4x16):  N = lane%16, v0/v1 = K (same half split)
//   C/D (16x16):    N = lane%16, VGPR d -> M = d (lanes 0-15) / 8+d (16-31)
// ---------------------------------------------------------------------------
template<int K>
__global__ void gemm_wmma_f32(const float* __restrict__ A, int lda,
                              const float* __restrict__ B, int ldb,
                              const float* __restrict__ bias,
                              const float* __restrict__ Cadd, int ldc,
                              float* __restrict__ D, int ldd,
                              int M, int N, int relu)
{
    __shared__ float sB[K * 16];               // fragment-swizzled (K x 16) strip

    const int tid  = threadIdx.x;
    const int lane = tid & 31;
    const int wid  = tid >> 5;
    const int mblocks = M >> 8;                // 256 rows per block
    const int mblk = blockIdx.x % mblocks;
    const int n0   = (blockIdx.x / mblocks) << 4;

    // sB[kt*64 + ln*2 + v] = B[kt*4 + (ln>>4)*2 + v][n0 + (ln&15)]
    for (int idx = tid; idx < K * 16; idx += 256) {
        const int kt = idx >> 6;
        const int r  = idx & 63;
        const int ln = r >> 1;
        const int v  = r & 1;
        const int kk = kt * 4 + ((ln >> 4) << 1) + v;
        sB[idx] = B[kk * ldb + n0 + (ln & 15)];
    }
    __syncthreads();

    const int row   = lane & 15;
    const int khalf = (lane >> 4) << 1;        // 0 or 2
    const int m0    = (mblk << 8) + (wid << 5);
    const float* A0 = A + (m0 + row) * lda + khalf;
    const float* A1 = A0 + (lda << 4);
    const v2f* sBv  = (const v2f*)sB;

    v8f acc0 = {0.f, 0.f, 0.f, 0.f, 0.f, 0.f, 0.f, 0.f};
    v8f acc1 = acc0;
#pragma unroll
    for (int kt = 0; kt < K / 4; ++kt) {
        v2f b  = sBv[(kt << 5) + lane];        // ds_load_b64
        v2f a0; a0.x = A0[kt * 4]; a0.y = A0[kt * 4 + 1];   // global_load_b64
        v2f a1; a1.x = A1[kt * 4]; a1.y = A1[kt * 4 + 1];
        acc0 = __builtin_amdgcn_wmma_f32_16x16x4_f32(false, a0, false, b,
                                                     (short)0, acc0, false, false);
        acc1 = __builtin_amdgcn_wmma_f32_16x16x4_f32(false, a1, false, b,
                                                     (short)0, acc1, false, false);
    }

    const int col = n0 + row;
    const float bv = bias ? bias[col] : 0.f;
#pragma unroll
    for (int d = 0; d < 8; ++d) {
        const int mr0 = m0 + ((lane < 16) ? d : 8 + d);
        const int mr1 = mr0 + 16;
        float v0 = acc0[d] + bv;
        float v1 = acc1[d] + bv;
        if (Cadd) { v0 += Cadd[mr0 * ldc + col]; v1 += Cadd[mr1 * ldc + col]; }
        if (relu) { v0 = fmaxf(v0, 0.f); v1 = fmaxf(v1, 0.f); }
        D[mr0 * ldd + col] = v0;
        D[mr1 * ldd + col] = v1;
    }
}

// ---------------------------------------------------------------------------
// Input projection: h[bt,n,:] = x[b,n,t,:45] @ Wp + bp,  bt = b*T + t
// ---------------------------------------------------------------------------
__global__ void proj_kernel(const float* __restrict__ x,
                            const float* __restrict__ Wp,
                            const float* __restrict__ bp,
                            float* __restrict__ h)
{
    const int n  = blockIdx.x & (GNN_N - 1);
    const int bt = blockIdx.x >> 7;
    const int b  = bt >> 2, t = bt & 3;
    const int f  = threadIdx.x;            // 64 threads
    const float* xr = x + (((b * GNN_N + n) * GNN_T + t) * GNN_IN);
    float acc = bp[f];
#pragma unroll
    for (int k = 0; k < GNN_IN; ++k) acc = fmaf(xr[k], Wp[k * GNN_F + f], acc);
    h[(bt * GNN_N + n) * GNN_F + f] = acc;
}

// ---------------------------------------------------------------------------
// Pairwise attention + aggregation (pre-Wm2), online softmax.
// One wave per (graph bt, node i); lane owns 4 contiguous H channels.
// bm/ba tiles (16 rows) DMA'd into LDS with async-to-LDS, reused by 8 waves.
//   r[i] = (1/s) * sum_{valid j} exp(logit_ij - m) * relu(am_i + bm_j)
//   logit_ij = relu(aa_i + ba_j) . wa2 + ba2
// ---------------------------------------------------------------------------
__global__ void pair_agg_kernel(const float* __restrict__ am,
                                const float* __restrict__ bm,
                                const float* __restrict__ aa,
                                const float* __restrict__ ba,
                                const float* __restrict__ wa2,
                                const float* __restrict__ ba2,
                                const float* __restrict__ masks,
                                float* __restrict__ r)
{
    __shared__ float s_bm[16 * GNN_H];
    __shared__ float s_ba[16 * GNN_H];
    __shared__ float s_mask[GNN_N];

    const int tid  = threadIdx.x;
    const int lane = tid & 31;
    const int w    = tid >> 5;                      // 8 waves
    const int bt   = blockIdx.x >> 4;               // 512 blocks total
    const int i    = ((blockIdx.x & 15) << 3) + w;
    const int b    = bt >> 2;
    const int base = bt * GNN_N;                    // row base in (4096,128)
    const int c    = lane << 2;                     // 4 channels / lane

    if (tid < GNN_N) s_mask[tid] = masks[b * GNN_N + tid];

    const uint32_t lbm = (uint32_t)(uintptr_t)s_bm;
    const uint32_t lba = (uint32_t)(uintptr_t)s_ba;

    const float4 ami = *(const float4*)(am + (base + i) * GNN_H + c);
    const float4 aai = *(const float4*)(aa + (base + i) * GNN_H + c);
    const float4 w2  = *(const float4*)(wa2 + c);
    const float  bb2 = ba2[0];

    float4 racc = {0.f, 0.f, 0.f, 0.f};
    float  mrun = -3.0e38f, srun = 0.f;

    for (int j0 = 0; j0 < GNN_N; j0 += 16) {
        __syncthreads();                            // previous tile consumed
        // async DMA stage: 16 rows x 128 f32 = 512 float4 per buffer
        for (int idx = tid; idx < 512; idx += 256) {
            async_copy_b128(lbm + idx * 16, bm + (base + j0) * GNN_H + idx * 4);
            async_copy_b128(lba + idx * 16, ba + (base + j0) * GNN_H + idx * 4);
        }
        wait_asynccnt0();
        __syncthreads();

        for (int jj = 0; jj < 16; ++jj) {
            const int j = j0 + jj;
            if (s_mask[j] == 0.f) continue;         // wave-uniform branch

            const float4 baj = *(const float4*)(s_ba + jj * GNN_H + c);
            float p = fmaxf(aai.x + baj.x, 0.f) * w2.x
                    + fmaxf(aai.y + baj.y, 0.f) * w2.y
                    + fmaxf(aai.z + baj.z, 0.f) * w2.z
                    + fmaxf(aai.w + baj.w, 0.f) * w2.w;
#pragma unroll
            for (int off = 16; off > 0; off >>= 1) p += __shfl_xor(p, off, 32);
            const float logit = p + bb2;

            const float nm    = fmaxf(mrun, logit);
            const float scale = __expf(mrun - nm);
            const float wgt   = __expf(logit - nm);

            const float4 bmj = *(const float4*)(s_bm + jj * GNN_H + c);
            racc.x = racc.x * scale + wgt * fmaxf(ami.x + bmj.x, 0.f);
            racc.y = racc.y * scale + wgt * fmaxf(ami.y + bmj.y, 0.f);
            racc.z = racc.z * scale + wgt * fmaxf(ami.z + bmj.z, 0.f);
            racc.w = racc.w * scale + wgt * fmaxf(ami.w + bmj.w, 0.f);
            srun   = srun * scale + wgt;
            mrun   = nm;
        }
    }
    const float inv = 1.f / srun;
    float4 outv = {racc.x * inv, racc.y * inv, racc.z * inv, racc.w * inv};
    *(float4*)(r + (base + i) * GNN_H + c) = outv;
}

// ---------------------------------------------------------------------------
// (BT,N,F) node-major  ->  (B,N,T,F) output layout
// ---------------------------------------------------------------------------
__global__ void untranspose_kernel(const float* __restrict__ hout,
                                   float* __restrict__ out)
{
    const int idx = blockIdx.x * blockDim.x + threadIdx.x;
    const int f = idx & 63;
    const int t = (idx >> 6) & 3;
    const int n = (idx >> 8) & 127;
    const int b = idx >> 15;
    out[idx] = hout[(((b << 2) + t) * GNN_N + n) * GNN_F + f];
}

// ---------------------------------------------------------------------------
static void launch_gemm(hipStream_t s, const float* A, int lda,
                        const float* B, int ldb, const float* bias,
                        const float* Cadd, int ldc, float* D, int ldd,
                        int M, int N, int K, int relu)
{
    const int blocks = (M / 256) * (N / 16);
    if (K == 64)
        gemm_wmma_f32<64><<<blocks, 256, 0, s>>>(A, lda, B, ldb, bias, Cadd, ldc,
                                                 D, ldd, M, N, relu);
    else
        gemm_wmma_f32<128><<<blocks, 256, 0, s>>>(A, lda, B, ldb, bias, Cadd, ldc,
                                                  D, ldd, M, N, relu);
}

extern "C" void kernel_launch(void* const* d_in, const int* in_sizes, int n_in,
                              void* d_out, int out_size, void* d_ws, size_t ws_size,
                              hipStream_t stream)
{
    const float* x     = (const float*)d_in[0];
    const float* masks = (const float*)d_in[1];
    const float* Wp    = (const float*)d_in[2];
    const float* bp    = (const float*)d_in[3];
    const float* Wm1   = (const float*)d_in[4];
    const float* bm1   = (const float*)d_in[5];
    const float* Wm2   = (const float*)d_in[6];
    const float* bm2   = (const float*)d_in[7];
    const float* Wa1   = (const float*)d_in[8];
    const float* ba1   = (const float*)d_in[9];
    const float* Wa2   = (const float*)d_in[10];
    const float* ba2   = (const float*)d_in[11];
    const float* Wu1   = (const float*)d_in[12];
    const float* bu1   = (const float*)d_in[13];
    const float* Wu2   = (const float*)d_in[14];
    const float* bu2   = (const float*)d_in[15];
    const float* Wo    = (const float*)d_in[16];
    const float* bo    = (const float*)d_in[17];
    float* out = (float*)d_out;

    // Workspace layout (floats). ~11.5 MB total with aliasing.
    float* ws = (float*)d_ws;
    float* h   = ws;                         // 4096*64
    float* am  = h  + GNN_M * GNN_F;         // 4096*128
    float* bmb = am + GNN_M * GNN_H;
    float* aab = bmb + GNN_M * GNN_H;
    float* bab = aab + GNN_M * GNN_H;
    float* rb  = bab + GNN_M * GNN_H;
    float* agg  = am;                        // dead after pair_agg
    float* tb   = bmb;
    float* hout = aab;

    // h = x @ Wp + bp  (folded to (BT,N,F) node-major)
    proj_kernel<<<GNN_M, GNN_F, 0, stream>>>(x, Wp, bp, h);

    for (int l = 0; l < 3; ++l) {
        const float* Wm1l = Wm1 + l * (2 * GNN_F) * GNN_H;
        const float* Wa1l = Wa1 + l * (2 * GNN_F) * GNN_H;
        const float* Wu1l = Wu1 + l * (GNN_F + GNN_H) * GNN_H;

        // per-node halves of the pair MLPs
        launch_gemm(stream, h, GNN_F, Wm1l,               GNN_H, bm1 + l*GNN_H, nullptr, 0, am,  GNN_H, GNN_M, GNN_H, GNN_F, 0);
        launch_gemm(stream, h, GNN_F, Wm1l + GNN_F*GNN_H, GNN_H, nullptr,       nullptr, 0, bmb, GNN_H, GNN_M, GNN_H, GNN_F, 0);
        launch_gemm(stream, h, GNN_F, Wa1l,               GNN_H, ba1 + l*GNN_H, nullptr, 0, aab, GNN_H, GNN_M, GNN_H, GNN_F, 0);
        launch_gemm(stream, h, GNN_F, Wa1l + GNN_F*GNN_H, GNN_H, nullptr,       nullptr, 0, bab, GNN_H, GNN_M, GNN_H, GNN_F, 0);

        // pairwise: masked online softmax + weighted relu-sum (pre-Wm2)
        pair_agg_kernel<<<GNN_BT * (GNN_N / 8), 256, 0, stream>>>(
            am, bmb, aab, bab, Wa2 + l * GNN_H, ba2 + l, masks, rb);

        // agg = r @ Wm2 + bm2   (Wm2 commuted past the aggregation)
        launch_gemm(stream, rb, GNN_H, Wm2 + l*GNN_H*GNN_H, GNN_H, bm2 + l*GNN_H, nullptr, 0, agg, GNN_H, GNN_M, GNN_H, GNN_H, 0);

        // upd = relu(h@Wu1_top + agg@Wu1_bot + bu1) @ Wu2 + bu2 ;  h += upd
        launch_gemm(stream, h,   GNN_F, Wu1l,               GNN_H, bu1 + l*GNN_H, nullptr, 0,      tb, GNN_H, GNN_M, GNN_H, GNN_F, 0);
        launch_gemm(stream, agg, GNN_H, Wu1l + GNN_F*GNN_H, GNN_H, nullptr,       tb, GNN_H,       tb, GNN_H, GNN_M, GNN_H, GNN_H, 1);
        launch_gemm(stream, tb,  GNN_H, Wu2 + l*GNN_H*GNN_F, GNN_F, bu2 + l*GNN_F, h, GNN_F,       h,  GNN_F, GNN_M, GNN_F, GNN_H, 0);
    }

    // out = h @ Wo + bo, then restore (B,N,T,F) layout
    launch_gemm(stream, h, GNN_F, Wo, GNN_F, bo, nullptr, 0, hout, GNN_F, GNN_M, GNN_F, GNN_F, 0);
    untranspose_kernel<<<(GNN_B*GNN_N*GNN_T*GNN_F) / 256, 256, 0, stream>>>(hout, out);
}